// Selector_8727373546119
// MI455X (gfx1250) — compile-verified
//
#include <hip/hip_runtime.h>
#include <hip/hip_bf16.h>

typedef __attribute__((ext_vector_type(2))) float v2f;
typedef __attribute__((ext_vector_type(8))) float v8f;
typedef unsigned long long u64;

#define EPSF 1e-8f
#define BQ   64     // queries
#define DD   256    // embed dim
#define NT   32     // K rows per block-iteration

// order-preserving float -> uint32 map (for packed max-selection)
__device__ __forceinline__ unsigned fmap(float s) {
    unsigned u = __float_as_uint(s);
    return (u & 0x80000000u) ? ~u : (u | 0x80000000u);
}

// register-resident sorted (ascending) top-32 list; c[0] is the min.
__device__ __forceinline__ void topk_insert(u64 (&c)[32], u64 p) {
    if (p <= c[0]) return;
    c[0] = p;
#pragma unroll
    for (int i = 0; i < 31; ++i) {
        u64 a = c[i], b = c[i + 1];
        u64 mn = a < b ? a : b;
        u64 mx = a < b ? b : a;
        c[i] = mn; c[i + 1] = mx;
    }
}

__device__ __forceinline__ u64 wavemax32(u64 v) {
#pragma unroll
    for (int m = 16; m >= 1; m >>= 1) {
        u64 o = __shfl_xor(v, m, 32);
        v = (o > v) ? o : v;
    }
    return v;
}

__device__ __forceinline__ void cswap_desc(u64 &a, u64 &b) {
    if (b > a) { u64 t = a; a = b; b = t; }
}

// Issue one 32-row x 256-col f32 tile as async memory->LDS b128 transfers.
// Each of the 256 threads issues 8 x 16B (tile = 32 KB).  Rows >= N skipped.
__device__ __forceinline__ void issue_tile_async(const float* __restrict__ ke,
                                                 int n0, int N, float* buf, int t) {
    unsigned lds_base = (unsigned)(uintptr_t)buf;   // low 32 bits = LDS byte addr
#pragma unroll
    for (int j = 0; j < 8; ++j) {
        int i = t + 256 * j;          // float4 index within tile
        int r = i >> 6;               // row 0..31
        int n = n0 + r;
        if (n < N) {
            int voff = n * (DD * 4) + (i & 63) * 16;   // byte offset (< 2^31)
            unsigned ldso = lds_base + (unsigned)i * 16u;
            asm volatile("global_load_async_to_lds_b128 %0, %1, %2 offset:0"
                         :: "v"(ldso), "v"(voff), "s"(ke)
                         : "memory");
        }
    }
}

// ---------------------------------------------------------------------------
// Pass 1: double-buffered async K-tile streaming + WMMA similarity +
// per-query top-32 within the block's N-range.
// ws[(q*S + b)*32 + rank] = packed candidate (desc).
// ---------------------------------------------------------------------------
__global__ __launch_bounds__(256)
void sel_pass1(const float* __restrict__ qg, const float* __restrict__ ke,
               u64* __restrict__ ws, int N, int rpb, int S) {
    __shared__ __align__(16) float lqs[BQ * DD];      // normalized queries (64 KB)
    __shared__ __align__(16) float lkb[2][NT * DD];   // double-buffered K tile (64 KB)
    __shared__ float lstage[8 * 256];                 // per-wave 16x16 transpose
    __shared__ float lrnorm[NT];
    __shared__ float lpart[256];
    __shared__ float lqn[BQ];

    const int t    = threadIdx.x;
    const int w    = t >> 5;
    const int lane = t & 31;

    const int nstart = blockIdx.x * rpb;
    const int iters  = rpb >> 5;

    // kick off tile 0 while we normalize the queries
    issue_tile_async(ke, nstart, N, lkb[0], t);

    // ---- load + normalize queries ----
    {
        const float4* gq = (const float4*)qg;
        float4* sq = (float4*)lqs;
#pragma unroll
        for (int i = t; i < (BQ * DD) / 4; i += 256) sq[i] = gq[i];
        __syncthreads();
        int r = t >> 2, seg = (t & 3) * 64;
        float ss = 0.f;
        const float4* q4 = (const float4*)&lqs[r * DD + seg];
#pragma unroll
        for (int k = 0; k < 16; ++k) {
            float4 v = q4[k];
            ss += v.x * v.x + v.y * v.y + v.z * v.z + v.w * v.w;
        }
        lpart[t] = ss;
        __syncthreads();
        if (t < BQ) {
            float s = lpart[t * 4] + lpart[t * 4 + 1] + lpart[t * 4 + 2] + lpart[t * 4 + 3];
            lqn[t] = 1.0f / fmaxf(sqrtf(s), EPSF);
        }
        __syncthreads();
        for (int i = t; i < BQ * DD; i += 256) lqs[i] *= lqn[i >> 8];
        // (tile-0 arrival barrier below also covers this write)
    }

    u64 cand[32];
#pragma unroll
    for (int i = 0; i < 32; ++i) cand[i] = 0ull;

    const int mt = w & 3;          // M tile (16 queries)
    const int ns = w >> 2;         // N subtile (16 rows)
    const float* aBase = &lqs[(mt * 16 + (lane & 15)) * DD + 2 * (lane >> 4)];
    const int    bOff  = (ns * 16 + (lane & 15)) * DD + 2 * (lane >> 4);
    float* st = &lstage[w * 256];

    for (int it = 0; it < iters; ++it) {
        const int n0 = nstart + it * NT;
        float* cur = lkb[it & 1];
        float* nxt = lkb[(it & 1) ^ 1];

        // tile `it` fully in LDS (own asyncs done, then block-wide barrier,
        // which also proves every wave finished reading `nxt` = tile it-1)
        asm volatile("s_wait_asynccnt 0" ::: "memory");
        __syncthreads();

        // prefetch tile it+1 into the other buffer (async, overlaps compute)
        if (it + 1 < iters)
            issue_tile_async(ke, n0 + NT, N, nxt, t);

        // ---- per-row inverse norms (float4 reads) ----
        {
            int r = t >> 3, seg = (t & 7) * 32;
            const float4* c4 = (const float4*)(cur + r * DD + seg);
            float ss = 0.f;
#pragma unroll
            for (int k = 0; k < 8; ++k) {
                float4 v = c4[k];
                ss += v.x * v.x + v.y * v.y + v.z * v.z + v.w * v.w;
            }
            lpart[t] = ss;
            __syncthreads();
            if (t < NT) {
                float s = 0.f;
#pragma unroll
                for (int j = 0; j < 8; ++j) s += lpart[t * 8 + j];
                lrnorm[t] = 1.0f / fmaxf(sqrtf(s), EPSF);
            }
        }
        __syncthreads();

        // ---- 16x16 tile via 64 chained V_WMMA_F32_16X16X4_F32 ----
        const float* bBase = cur + bOff;
        v8f acc = {};
#pragma unroll 4
        for (int k = 0; k < DD; k += 4) {
            v2f a = *(const v2f*)(aBase + k);
            v2f b = *(const v2f*)(bBase + k);
            acc = __builtin_amdgcn_wmma_f32_16x16x4_f32(
                false, a, false, b, (short)0, acc, false, false);
        }

        // ---- scale by row norm, transpose through per-wave LDS ----
        const float rn = lrnorm[ns * 16 + (lane & 15)];
        const int nl = lane & 15;
        const int mh = (lane >> 4) * 8;
#pragma unroll
        for (int r = 0; r < 8; ++r)
            st[(mh + r) * 16 + nl] = acc[r] * rn;
        asm volatile("s_wait_dscnt 0" ::: "memory");   // wave-local ordering

        const int qloc  = lane & 15;
        const int h     = lane >> 4;
        const int nbase = n0 + ns * 16 + h * 8;
#pragma unroll
        for (int j = 0; j < 8; ++j) {
            float s = st[qloc * 16 + h * 8 + j];
            int n = nbase + j;
            u64 p = (n < N)
                  ? ((((u64)fmap(s)) << 32) | (unsigned)~(unsigned)n)
                  : 0ull;
            topk_insert(cand, p);
        }
        asm volatile("s_wait_dscnt 0" ::: "memory");   // reads done before rewrite
    }

    // ---- block epilogue: 4 partitions x top-32 per query -> top-32 ----
    __syncthreads();
    u64* lcand = (u64*)lqs;   // reuse 64 KB: 64 queries x 128 candidates
    {
        int q    = (w & 3) * 16 + (lane & 15);
        int slot = ((w >> 2) << 1) + (lane >> 4);
#pragma unroll
        for (int i = 0; i < 32; ++i)
            lcand[q * 128 + slot * 32 + i] = cand[31 - i];   // descending
    }
    __syncthreads();

    // each wave merges 8 queries via 32-round k-way merge over lane lists
    for (int qq = w * 8; qq < w * 8 + 8; ++qq) {
        u64 s0 = lcand[qq * 128 + lane * 4 + 0];
        u64 s1 = lcand[qq * 128 + lane * 4 + 1];
        u64 s2 = lcand[qq * 128 + lane * 4 + 2];
        u64 s3 = lcand[qq * 128 + lane * 4 + 3];
        cswap_desc(s0, s1); cswap_desc(s2, s3);
        cswap_desc(s0, s2); cswap_desc(s1, s3); cswap_desc(s1, s2);
        u64 res = 0;
#pragma unroll 1
        for (int r = 0; r < 32; ++r) {
            u64 h = s0;
            u64 m = wavemax32(h);
            if (lane == r) res = m;
            u64 ball = __ballot(h == m);
            int first = __ffsll((long long)ball) - 1;
            if (lane == first) { s0 = s1; s1 = s2; s2 = s3; s3 = 0; }
        }
        ws[((size_t)qq * S + blockIdx.x) * 32 + lane] = res;
    }
}

// ---------------------------------------------------------------------------
// Pass 2: one block per query; merge S*32 candidates -> top-32, emit ids and
// gathered embeddings.  out[0:2048] = ids (float), out[2048:] = [64,32,256].
// ---------------------------------------------------------------------------
__global__ __launch_bounds__(256)
void sel_pass2(const u64* __restrict__ ws, const float* __restrict__ ke,
               const int* __restrict__ kf, float* __restrict__ out,
               int S, int N) {
    __shared__ u64 buf[256 * 32];   // 64 KB
    __shared__ u64 red[256];
    __shared__ u64 top[32];

    const int q = blockIdx.x;
    const int t = threadIdx.x;
    const int total = S * 32;

    u64 c[32];
#pragma unroll
    for (int i = 0; i < 32; ++i) c[i] = 0ull;
    const u64* src = ws + (size_t)q * total;
    for (int i = t; i < total; i += 256) topk_insert(c, src[i]);
#pragma unroll
    for (int i = 0; i < 32; ++i) buf[t * 32 + i] = c[i];
    __syncthreads();

    for (int r = 0; r < 32; ++r) {
        u64 lm = 0;
#pragma unroll
        for (int i = 0; i < 32; ++i) { u64 v = buf[t * 32 + i]; lm = v > lm ? v : lm; }
        red[t] = lm;
        __syncthreads();
        for (int s = 128; s > 0; s >>= 1) {
            if (t < s) { u64 a = red[t], b = red[t + s]; red[t] = b > a ? b : a; }
            __syncthreads();
        }
        u64 M = red[0];
#pragma unroll
        for (int i = 0; i < 32; ++i)
            if (buf[t * 32 + i] == M) buf[t * 32 + i] = 0ull;
        if (t == 0) top[r] = M;
        __syncthreads();
    }

    // ids
    if (t < 32) {
        unsigned n = ~(unsigned)(top[t] & 0xffffffffull);
        if (n >= (unsigned)N) n = 0;
        out[q * 32 + t] = (float)kf[n];
    }
    __syncthreads();
    // embedding gather: 32 rows x 256 floats
    for (int j = 0; j < 32; ++j) {
        unsigned n = ~(unsigned)(top[j] & 0xffffffffull);
        if (n >= (unsigned)N) n = 0;
        out[2048 + ((size_t)q * 32 + j) * DD + t] = ke[(size_t)n * DD + t];
    }
}

extern "C" void kernel_launch(void* const* d_in, const int* in_sizes, int n_in,
                              void* d_out, int out_size, void* d_ws, size_t ws_size,
                              hipStream_t stream) {
    const float* qg = (const float*)d_in[0];
    const float* ke = (const float*)d_in[1];
    const int*   kf = (const int*)d_in[2];
    float* out = (float*)d_out;

    const int N = in_sizes[1] / DD;   // 1,000,000

    // choose split count from available workspace: each block needs 64*32*8 B
    size_t per_block = (size_t)BQ * 32 * sizeof(u64);   // 16 KB
    size_t smax = ws_size / per_block;
    int S = (int)(smax < 512 ? smax : 512);
    if (S < 1) S = 1;
    int rpb = (((N + S - 1) / S) + 31) & ~31;   // rows per block, multiple of 32

    sel_pass1<<<S, 256, 0, stream>>>(qg, ke, (u64*)d_ws, N, rpb, S);
    sel_pass2<<<BQ, 256, 0, stream>>>((const u64*)d_ws, ke, kf, out, S, N);
}